// STGNN_56221121905004
// MI455X (gfx1250) — compile-verified
//
#include <hip/hip_runtime.h>
#include <math.h>

typedef __attribute__((ext_vector_type(2))) float v2f;
typedef __attribute__((ext_vector_type(8))) float v8f;

__device__ __forceinline__ v8f wmma4(v2f a, v2f b, v8f c) {
  // D = A(16x4 f32) * B(4x16 f32) + C(16x16 f32)
  return __builtin_amdgcn_wmma_f32_16x16x4_f32(
      /*neg_a=*/false, a, /*neg_b=*/false, b,
      /*c_mod=*/(short)0, c, /*reuse_a=*/false, /*reuse_b=*/false);
}

// ---------------- scratch zeroing ----------------
__global__ void k_zero(float* __restrict__ p, long long n) {
  long long i = (long long)blockIdx.x * blockDim.x + threadIdx.x;
  long long stride = (long long)gridDim.x * blockDim.x;
  for (; i < n; i += stride) p[i] = 0.0f;
}

// ---------------- degree histograms ----------------
__global__ void k_deg(const int* __restrict__ src, const int* __restrict__ dst,
                      float* __restrict__ deg_src, float* __restrict__ deg_dst, int E) {
  int e = blockIdx.x * blockDim.x + threadIdx.x;
  if (e >= E) return;
  atomicAdd(&deg_src[src[e]], 1.0f);
  atomicAdd(&deg_dst[dst[e]], 1.0f);
}

// ---------------- normalizations ----------------
__global__ void k_dis(const float* __restrict__ deg_src, const float* __restrict__ deg_dst,
                      float* __restrict__ dis, float* __restrict__ disg, int n) {
  int i = blockIdx.x * blockDim.x + threadIdx.x;
  if (i >= n) return;
  float d = deg_src[i];
  dis[i]  = (d > 0.0f) ? rsqrtf(d) : 0.0f;   // deg integer => max(deg,1)=deg when deg>0
  disg[i] = rsqrtf(deg_dst[i] + 1.0f);
}

// ---------------- weighted edge scatter: out[dst] += sgn*scale[s]*scale[d] * v[src] ----------------
// 8 threads per edge, 4 floats each (full 32-feature row, float4 gathers).
__global__ void k_scatter(const int* __restrict__ src, const int* __restrict__ dst,
                          const float* __restrict__ v, const float* __restrict__ scale,
                          float sgn, float* __restrict__ out, int E) {
  long long t = (long long)blockIdx.x * blockDim.x + threadIdx.x;
  long long e = t >> 3;
  int c = (int)(t & 7) * 4;
  if (e >= (long long)E) return;
  int s = src[e], d = dst[e];
  float w = sgn * scale[s] * scale[d];
  const float4 xv = *(const float4*)(v + (long long)s * 32 + c);
  float* o = out + (long long)d * 32 + c;
  atomicAdd(o + 0, w * xv.x);
  atomicAdd(o + 1, w * xv.y);
  atomicAdd(o + 2, w * xv.z);
  atomicAdd(o + 3, w * xv.w);
}

// ---------------- fused gates + hw = ((1-Z)*tanh(..)) @ Wgcn, WMMA f32 16x16x4 ----------------
// One wave per 16-row node tile; 8 waves (128 rows) per block. Requires n % 16 == 0.
__global__ void __launch_bounds__(256)
k_gates_wmma(const float* __restrict__ x, const float* __restrict__ Lx,
             const float* __restrict__ Wxz, const float* __restrict__ Wxh,
             const float* __restrict__ Wgcn,
             const float* __restrict__ bxz, const float* __restrict__ bhz,
             const float* __restrict__ bxh, const float* __restrict__ bhh,
             float* __restrict__ hw, int n) {
  __shared__ float lds[8 * 16 * 33];            // per-wave 16x32 H tile, stride 33 (bank-conflict pad)
  const int lane = threadIdx.x & 31;
  const int wave = threadIdx.x >> 5;
  const int half = lane >> 4;                   // 0: K pair {0,1}, 1: K pair {2,3} (A/B operand split)
  const int lid  = lane & 15;
  const int row_base = blockIdx.x * 128 + wave * 16;
  if (row_base >= n) return;                    // wave-uniform: EXEC stays all-ones for WMMA
  float* sm = lds + wave * (16 * 33);

  v8f cz0 = {}, cz1 = {}, ch0 = {}, ch1 = {};

  const float* arow_x  = x  + (size_t)(row_base + lid) * 32 + 2 * half;
  const float* arow_lx = Lx + (size_t)(row_base + lid) * 32 + 2 * half;
  const float* W0z = Wxz;            // W_xz[0]
  const float* W1z = Wxz + 1024;     // W_xz[1]
  const float* W0h = Wxh;
  const float* W1h = Wxh + 1024;

  #pragma unroll
  for (int kk = 0; kk < 32; kk += 4) {          // x @ W0z, x @ W0h
    v2f a; a.x = arow_x[kk]; a.y = arow_x[kk + 1];
    const float* wr = W0z + (kk + 2 * half) * 32;
    v2f b0, b1;
    b0.x = wr[lid];      b0.y = wr[32 + lid];
    b1.x = wr[16 + lid]; b1.y = wr[48 + lid];
    cz0 = wmma4(a, b0, cz0);
    cz1 = wmma4(a, b1, cz1);
    wr = W0h + (kk + 2 * half) * 32;
    b0.x = wr[lid];      b0.y = wr[32 + lid];
    b1.x = wr[16 + lid]; b1.y = wr[48 + lid];
    ch0 = wmma4(a, b0, ch0);
    ch1 = wmma4(a, b1, ch1);
  }
  #pragma unroll
  for (int kk = 0; kk < 32; kk += 4) {          // += Lx @ W1z, Lx @ W1h
    v2f a; a.x = arow_lx[kk]; a.y = arow_lx[kk + 1];
    const float* wr = W1z + (kk + 2 * half) * 32;
    v2f b0, b1;
    b0.x = wr[lid];      b0.y = wr[32 + lid];
    b1.x = wr[16 + lid]; b1.y = wr[48 + lid];
    cz0 = wmma4(a, b0, cz0);
    cz1 = wmma4(a, b1, cz1);
    wr = W1h + (kk + 2 * half) * 32;
    b0.x = wr[lid];      b0.y = wr[32 + lid];
    b1.x = wr[16 + lid]; b1.y = wr[48 + lid];
    ch0 = wmma4(a, b0, ch0);
    ch1 = wmma4(a, b1, ch1);
  }

  // gates; D layout: element v of lane = row (v + 8*half), col (lid | lid+16)
  const float bz0 = bxz[lid]      + bhz[lid];
  const float bz1 = bxz[16 + lid] + bhz[16 + lid];
  const float bh0 = bxh[lid]      + bhh[lid];
  const float bh1 = bxh[16 + lid] + bhh[16 + lid];
  #pragma unroll
  for (int v = 0; v < 8; ++v) {
    int m = v + 8 * half;
    float z0 = 1.0f / (1.0f + __expf(-(cz0[v] + bz0)));
    float z1 = 1.0f / (1.0f + __expf(-(cz1[v] + bz1)));
    float H0 = (1.0f - z0) * tanhf(ch0[v] + bh0);
    float H1 = (1.0f - z1) * tanhf(ch1[v] + bh1);
    sm[m * 33 + lid]      = H0;
    sm[m * 33 + 16 + lid] = H1;
  }
  // same-wave LDS store->load: DS ops are in-order per wave; compiler inserts s_wait_dscnt

  v8f cw0 = {}, cw1 = {};
  #pragma unroll
  for (int kk = 0; kk < 32; kk += 4) {          // hw = H @ Wgcn (A read back transposed from LDS)
    v2f a;
    a.x = sm[lid * 33 + kk + 2 * half];
    a.y = sm[lid * 33 + kk + 2 * half + 1];
    const float* wr = Wgcn + (kk + 2 * half) * 32;
    v2f b0, b1;
    b0.x = wr[lid];      b0.y = wr[32 + lid];
    b1.x = wr[16 + lid]; b1.y = wr[48 + lid];
    cw0 = wmma4(a, b0, cw0);
    cw1 = wmma4(a, b1, cw1);
  }
  #pragma unroll
  for (int v = 0; v < 8; ++v) {
    size_t m = (size_t)(row_base + v + 8 * half);
    hw[m * 32 + lid]      = cw0[v];
    hw[m * 32 + 16 + lid] = cw1[v];
  }
}

// ---------------- finalize: relu(agg + disg^2*hw + b_gcn) . W_lin + b_lin ----------------
__global__ void k_final(const float* __restrict__ aggg, const float* __restrict__ hwv,
                        const float* __restrict__ disg, const float* __restrict__ bgcn,
                        const float* __restrict__ Wlin, const float* __restrict__ blin,
                        float* __restrict__ out, int n) {
  int i = blockIdx.x * blockDim.x + threadIdx.x;
  if (i >= n) return;
  float ds = disg[i];
  float ds2 = ds * ds;
  const float* ag = aggg + (size_t)i * 32;
  const float* hv = hwv  + (size_t)i * 32;
  float acc = 0.0f;
  #pragma unroll
  for (int f = 0; f < 32; ++f) {
    float h = ag[f] + ds2 * hv[f] + bgcn[f];
    h = fmaxf(h, 0.0f);
    acc += h * Wlin[f];
  }
  out[i] = acc + blin[0];
}

extern "C" void kernel_launch(void* const* d_in, const int* in_sizes, int n_in,
                              void* d_out, int out_size, void* d_ws, size_t ws_size,
                              hipStream_t stream) {
  (void)n_in; (void)out_size; (void)ws_size;
  const float* x    = (const float*)d_in[0];
  const int*   ei   = (const int*)d_in[1];
  const int N = in_sizes[0] / 32;
  const int E = in_sizes[1] / 2;
  const int* src = ei;            // edge_index[0]
  const int* dst = ei + E;        // edge_index[1]
  const float* Wxz  = (const float*)d_in[2];
  const float* bxz  = (const float*)d_in[3];
  const float* bhz  = (const float*)d_in[5];
  // W_hz/W_xr/b_xr/W_hr/b_hr/W_hh unused: H0 == 0 collapses cheb(H0,.) to bias, R is dead
  const float* Wxh  = (const float*)d_in[10];
  const float* bxh  = (const float*)d_in[11];
  const float* bhh  = (const float*)d_in[13];
  const float* Wgcn = (const float*)d_in[14];
  const float* bgcn = (const float*)d_in[15];
  const float* Wlin = (const float*)d_in[16];
  const float* blin = (const float*)d_in[17];

  // workspace layout (floats): deg_src[N] deg_dst[N] Lx[32N] aggg[32N] dis[N] disg[N] hw[32N]
  float* ws = (float*)d_ws;
  const size_t Nl = (size_t)N;
  float* deg_src = ws;
  float* deg_dst = ws + Nl;
  float* Lx      = ws + 2 * Nl;
  float* aggg    = ws + 34 * Nl;
  float* dis     = ws + 66 * Nl;
  float* disg    = ws + 67 * Nl;
  float* hw      = ws + 68 * Nl;

  const long long zn = 66LL * (long long)N;     // regions needing accumulate-from-zero
  k_zero<<<2048, 256, 0, stream>>>(ws, zn);
  k_deg<<<(E + 255) / 256, 256, 0, stream>>>(src, dst, deg_src, deg_dst, E);
  k_dis<<<(N + 255) / 256, 256, 0, stream>>>(deg_src, deg_dst, dis, disg, N);

  const long long t8 = 8LL * (long long)E;
  k_scatter<<<(unsigned)((t8 + 255) / 256), 256, 0, stream>>>(src, dst, x, dis, -1.0f, Lx, E);

  k_gates_wmma<<<(N + 127) / 128, 256, 0, stream>>>(x, Lx, Wxz, Wxh, Wgcn,
                                                    bxz, bhz, bxh, bhh, hw, N);

  k_scatter<<<(unsigned)((t8 + 255) / 256), 256, 0, stream>>>(src, dst, hw, disg, 1.0f, aggg, E);

  k_final<<<(N + 255) / 256, 256, 0, stream>>>(aggg, hw, disg, bgcn, Wlin, blin,
                                               (float*)d_out, N);
}